// JKRGCN_34548716929227
// MI455X (gfx1250) — compile-verified
//
#include <hip/hip_runtime.h>
#include <hip/hip_bf16.h>

// ---------------------------------------------------------------------------
// JK-RGCN (paper/author/inst) on MI455X (gfx1250, wave32, WMMA)
//   - mean-SpMM: edge-parallel wave-per-edge scatter with f32 atomics,
//     degree-inverse folded into the scatter (HBM/atomic bound at 23.3 TB/s)
//   - dense GEMMs: V_WMMA_F32_16X16X4_F32, one wave per 16x16 C tile, K=128
//   - dead work removed: layer-2 author/inst branches + 'ai' relation unused
// ---------------------------------------------------------------------------

typedef __attribute__((ext_vector_type(2))) float v2f;
typedef __attribute__((ext_vector_type(8))) float v8f;

#define NP_N 200000
#define NA_N 100000
#define NI_N 10000
#define DIM 128

// ---------------- degree kernels ----------------
__global__ void count_deg_kernel(const int* __restrict__ dst, float* __restrict__ deg, int E) {
    int t = blockIdx.x * blockDim.x + threadIdx.x;
    if (t < E) atomicAdd(&deg[dst[t]], 1.0f);
}

__global__ void invert_deg_kernel(float* __restrict__ deg, int n) {
    int t = blockIdx.x * blockDim.x + threadIdx.x;
    if (t < n) deg[t] = 1.0f / fmaxf(deg[t], 1.0f);
}

// ---------------- mean-SpMM scatter: agg[dst] += x[src] * invdeg[dst] -------
// one wave (32 lanes) per edge, each lane moves 4 contiguous floats (128/32)
__global__ void spmm_scatter_kernel(const float* __restrict__ X,
                                    const int* __restrict__ src,
                                    const int* __restrict__ dst,
                                    const float* __restrict__ invdeg,
                                    float* __restrict__ agg, int E) {
    int t = blockIdx.x * blockDim.x + threadIdx.x;
    int e = t >> 5;
    if (e >= E) return;
    int lane = t & 31;
    int s = src[e], d = dst[e];
    float sc = invdeg[d];
    const float4 v = *((const float4*)(X + (size_t)s * DIM) + lane);
    float* ap = agg + (size_t)d * DIM + lane * 4;
    atomicAdd(ap + 0, v.x * sc);
    atomicAdd(ap + 1, v.y * sc);
    atomicAdd(ap + 2, v.z * sc);
    atomicAdd(ap + 3, v.w * sc);
}

// ---------------- WMMA K=128 accumulation helper ----------------------------
// A frag (16x4 f32): lane l holds rows r=l&15, K pair koff=(l>>4)*2
// B frag (4x16 f32): lane l holds col  c=l&15, K pair koff=(l>>4)*2
__device__ __forceinline__ v8f wmma_accum_k128(v8f acc,
                                               const float* __restrict__ X,
                                               const float* __restrict__ W,
                                               int row, int col, int koff) {
    const float* xr = X + (size_t)row * DIM;
#pragma unroll 8
    for (int k = 0; k < DIM; k += 4) {
        v2f a;
        a.x = xr[k + koff];
        a.y = xr[k + koff + 1];
        v2f b;
        b.x = W[(size_t)(k + koff) * DIM + col];
        b.y = W[(size_t)(k + koff + 1) * DIM + col];
        acc = __builtin_amdgcn_wmma_f32_16x16x4_f32(false, a, false, b,
                                                    (short)0, acc, false, false);
    }
    return acc;
}

// ---------------- fused multi-input GEMM + bias + ReLU ----------------------
// Y[N,128] = relu(X0@W0 + X1@W1 + X2@W2 + bias); X1/X2 may be nullptr.
// block = 128 threads (4 waves); wave w -> col tile (blockIdx.y*4+w)
// grid  = (N/16, 2)
__global__ void fused_gemm_relu_kernel(const float* __restrict__ X0, const float* __restrict__ W0,
                                       const float* __restrict__ X1, const float* __restrict__ W1,
                                       const float* __restrict__ X2, const float* __restrict__ W2,
                                       const float* __restrict__ bias,
                                       float* __restrict__ Y) {
    const int lane = threadIdx.x & 31;
    const int wave = threadIdx.x >> 5;
    const int rowbase = blockIdx.x * 16;
    const int colbase = (blockIdx.y * 4 + wave) * 16;
    const int r = lane & 15;
    const int koff = (lane >> 4) * 2;
    const int arow = rowbase + r;
    const int bcol = colbase + r;

    v8f acc = {};
    acc = wmma_accum_k128(acc, X0, W0, arow, bcol, koff);
    if (X1) acc = wmma_accum_k128(acc, X1, W1, arow, bcol, koff);
    if (X2) acc = wmma_accum_k128(acc, X2, W2, arow, bcol, koff);

    const float bv = bias[bcol];
    const int rowoff = (lane >> 4) * 8;
#pragma unroll
    for (int v = 0; v < 8; ++v) {
        float val = acc[v] + bv;
        val = val > 0.0f ? val : 0.0f;
        Y[(size_t)(rowbase + rowoff + v) * DIM + bcol] = val;
    }
}

// ---------------- head GEMM: [h1|h2] (K=256) @ out_w[256,40] + out_b --------
// one wave per 16x16 tile; grid = (NP/16, 3) -> 48 cols, stores masked to 40
__global__ void out_gemm_kernel(const float* __restrict__ H1, const float* __restrict__ H2,
                                const float* __restrict__ W, const float* __restrict__ B,
                                float* __restrict__ out) {
    const int lane = threadIdx.x & 31;
    const int rowbase = blockIdx.x * 16;
    const int colbase = blockIdx.y * 16;
    const int r = lane & 15;
    const int koff = (lane >> 4) * 2;
    const int col = colbase + r;
    const int colc = (col < 40) ? col : 0;   // keep B loads in bounds; junk cols dropped

    v8f acc = {};
    const float* h1r = H1 + (size_t)(rowbase + r) * DIM;
    const float* h2r = H2 + (size_t)(rowbase + r) * DIM;
#pragma unroll 8
    for (int k = 0; k < 256; k += 4) {
        const float* hr = (k < DIM) ? h1r : h2r;
        const int kk = (k < DIM) ? k : (k - DIM);
        v2f a;
        a.x = hr[kk + koff];
        a.y = hr[kk + koff + 1];
        v2f b;
        b.x = W[(size_t)(k + koff) * 40 + colc];
        b.y = W[(size_t)(k + koff + 1) * 40 + colc];
        acc = __builtin_amdgcn_wmma_f32_16x16x4_f32(false, a, false, b,
                                                    (short)0, acc, false, false);
    }
    if (col < 40) {
        const float bv = B[col];
        const int rowoff = (lane >> 4) * 8;
#pragma unroll
        for (int v = 0; v < 8; ++v)
            out[(size_t)(rowbase + rowoff + v) * 40 + col] = acc[v] + bv;
    }
}

__global__ void alpha_kernel(float* __restrict__ out) {
    if (threadIdx.x == 0 && blockIdx.x == 0) {
        out[(size_t)NP_N * 40 + 0] = 0.5f;
        out[(size_t)NP_N * 40 + 1] = 0.5f;
    }
}

// ---------------------------------------------------------------------------
extern "C" void kernel_launch(void* const* d_in, const int* in_sizes, int n_in,
                              void* d_out, int out_size, void* d_ws, size_t ws_size,
                              hipStream_t stream) {
    const float* x_paper    = (const float*)d_in[0];
    const float* emb_author = (const float*)d_in[1];
    // d_in[2] = emb_inst: dead (inst branch never reaches the output head)
    const float* w_pp = (const float*)d_in[3];   // [2,128,128]
    const float* w_ap = (const float*)d_in[4];
    const float* w_pa = (const float*)d_in[5];
    // d_in[6] = w_rel_ai: dead
    const float* w_rp = (const float*)d_in[7];
    const float* w_ra = (const float*)d_in[8];
    // d_in[9] = w_root_i: dead
    const float* b_rp = (const float*)d_in[10];  // [2,128]
    const float* b_ra = (const float*)d_in[11];
    // d_in[12] = b_root_i: dead
    const float* out_w = (const float*)d_in[13]; // [256,40]
    const float* out_b = (const float*)d_in[14]; // [40]
    const int* pp_src = (const int*)d_in[15];
    const int* pp_dst = (const int*)d_in[16];
    const int* ap_src = (const int*)d_in[17];
    const int* ap_dst = (const int*)d_in[18];
    const int* pa_src = (const int*)d_in[19];
    const int* pa_dst = (const int*)d_in[20];
    // d_in[21..22] = ai edges: dead
    const int E_pp = in_sizes[15];
    const int E_ap = in_sizes[17];
    const int E_pa = in_sizes[19];

    const size_t WOFF = (size_t)DIM * DIM; // per-layer weight stride

    // ---- workspace layout (floats) ----
    float* ws = (float*)d_ws;
    size_t off = 0;
    float* invdeg_pp = ws + off; off += NP_N;
    float* invdeg_ap = ws + off; off += NP_N;
    float* invdeg_pa = ws + off; off += NA_N;
    const size_t n_invdeg = off;
    float* agg_pp = ws + off; off += (size_t)NP_N * DIM;
    float* agg_ap = ws + off; off += (size_t)NP_N * DIM;
    float* agg_pa = ws + off; off += (size_t)NA_N * DIM;
    float* hp1 = ws + off; off += (size_t)NP_N * DIM;
    float* hp2 = ws + off; off += (size_t)NP_N * DIM;
    float* xa1 = ws + off; off += (size_t)NA_N * DIM;

    // ---- inverse degrees (edges are static but must be recomputed: stateless) ----
    hipMemsetAsync(invdeg_pp, 0, n_invdeg * sizeof(float), stream);
    count_deg_kernel<<<(E_pp + 255) / 256, 256, 0, stream>>>(pp_dst, invdeg_pp, E_pp);
    count_deg_kernel<<<(E_ap + 255) / 256, 256, 0, stream>>>(ap_dst, invdeg_ap, E_ap);
    count_deg_kernel<<<(E_pa + 255) / 256, 256, 0, stream>>>(pa_dst, invdeg_pa, E_pa);
    invert_deg_kernel<<<((int)n_invdeg + 255) / 256, 256, 0, stream>>>(invdeg_pp, (int)n_invdeg);

    // ---- layer 1 aggregations ----
    hipMemsetAsync(agg_pp, 0, ((size_t)NP_N + NP_N + NA_N) * DIM * sizeof(float), stream);
    spmm_scatter_kernel<<<(E_pp + 7) / 8, 256, 0, stream>>>(x_paper, pp_src, pp_dst, invdeg_pp, agg_pp, E_pp);
    spmm_scatter_kernel<<<(E_ap + 7) / 8, 256, 0, stream>>>(emb_author, ap_src, ap_dst, invdeg_ap, agg_ap, E_ap);
    spmm_scatter_kernel<<<(E_pa + 7) / 8, 256, 0, stream>>>(x_paper, pa_src, pa_dst, invdeg_pa, agg_pa, E_pa);

    // ---- layer 1 fused GEMMs (root + relation linears + bias + relu) ----
    fused_gemm_relu_kernel<<<dim3(NP_N / 16, 2), 128, 0, stream>>>(
        x_paper, w_rp, agg_pp, w_pp, agg_ap, w_ap, b_rp, hp1);
    fused_gemm_relu_kernel<<<dim3(NA_N / 16, 2), 128, 0, stream>>>(
        emb_author, w_ra, agg_pa, w_pa, nullptr, nullptr, b_ra, xa1);

    // ---- layer 2 aggregations (paper destinations only) ----
    hipMemsetAsync(agg_pp, 0, (size_t)2 * NP_N * DIM * sizeof(float), stream);
    spmm_scatter_kernel<<<(E_pp + 7) / 8, 256, 0, stream>>>(hp1, pp_src, pp_dst, invdeg_pp, agg_pp, E_pp);
    spmm_scatter_kernel<<<(E_ap + 7) / 8, 256, 0, stream>>>(xa1, ap_src, ap_dst, invdeg_ap, agg_ap, E_ap);

    fused_gemm_relu_kernel<<<dim3(NP_N / 16, 2), 128, 0, stream>>>(
        hp1, w_rp + WOFF, agg_pp, w_pp + WOFF, agg_ap, w_ap + WOFF, b_rp + DIM, hp2);

    // ---- head: logits = [hp1|hp2] @ out_w + out_b ; alpha = {0.5, 0.5} ----
    out_gemm_kernel<<<dim3(NP_N / 16, 3), 32, 0, stream>>>(hp1, hp2, out_w, out_b, (float*)d_out);
    alpha_kernel<<<1, 32, 0, stream>>>((float*)d_out);
}